// GraphSAGE_Net_69363721831026
// MI455X (gfx1250) — compile-verified
//
#include <hip/hip_runtime.h>
#include <math.h>

// GraphSAGE 3-layer (mean aggr) for MI455X / gfx1250.
// Scatter-bound: fp atomics for aggregation; fp32 WMMA (16x16x4) for GEMMs;
// TDM (tensor_load_to_lds) stages the 16x128 A-panels into LDS once per block.

#define N_NODES   50000
#define N_EDGES   800000
#define D_FEAT    128
#define HIDDEN    128
#define N_CLASSES 40
#define M_TILES   (N_NODES / 16)   // 3125, divides exactly -> no row masking

typedef __attribute__((ext_vector_type(2))) float v2f;
typedef __attribute__((ext_vector_type(8))) float v8f;
typedef __attribute__((ext_vector_type(4))) unsigned int u32x4;
typedef __attribute__((ext_vector_type(8))) int i32x8;
typedef __attribute__((ext_vector_type(4))) int i32x4;

// ---------------------------------------------------------------- utilities
__global__ void k_zero(float* __restrict__ p, int n) {
  int i = blockIdx.x * blockDim.x + threadIdx.x;
  if (i < n) p[i] = 0.0f;
}

__global__ void k_degree(const long long* __restrict__ dst, float* __restrict__ deg) {
  int e = blockIdx.x * blockDim.x + threadIdx.x;
  if (e < N_EDGES) atomicAdd(&deg[(int)dst[e]], 1.0f);
}

__global__ void k_invdeg(float* __restrict__ deg) {
  int i = blockIdx.x * blockDim.x + threadIdx.x;
  if (i < N_NODES) deg[i] = 1.0f / fmaxf(deg[i], 1.0f);
}

// 32 lanes per edge; each lane moves a float4 of the 128 features.
// Gather x[src] (coalesced 512B/edge) + 4x GLOBAL_ATOMIC_ADD_F32 into agg[dst].
__global__ void k_scatter(const float* __restrict__ x,
                          const long long* __restrict__ src,
                          const long long* __restrict__ dst,
                          float* __restrict__ agg) {
  int t = blockIdx.x * blockDim.x + threadIdx.x;
  int e = t >> 5;
  if (e >= N_EDGES) return;
  int g = (t & 31) << 2;
  int s = (int)src[e];
  int d = (int)dst[e];
  const float4 v = *(const float4*)(x + (size_t)s * D_FEAT + g);
  float* o = agg + (size_t)d * D_FEAT + g;
  atomicAdd(o + 0, v.x);
  atomicAdd(o + 1, v.y);
  atomicAdd(o + 2, v.z);
  atomicAdd(o + 3, v.w);
}

// ------------------------------------------------- TDM 1-D panel load to LDS
// Minimal D# for a contiguous run of nElems 4-byte elements (ISA 08 §8.3/8.4):
//   group0: count=1 | lds_addr | global_addr[56:0] | type=2
//   group1: data_size=4B; tensor_dim0 = tile_dim0 = nElems; dim0_stride = nElems;
//           tensor_dim1 = 1, tile_dim1/2 = 0 (1-D); no pad/iterate/multicast.
//   groups 2/3 + trailing group: zero (<=2-D tensor).
// This toolchain exposes the 6-arg builtin:
//   (uint32x4 g0, int32x8 g1, int32x4 g2, int32x4 g3, int32x8 extra, i32 cpol)
#if __has_builtin(__builtin_amdgcn_tensor_load_to_lds)
#define HAVE_TDM 1
__device__ __forceinline__ void tdm_load_1d(unsigned lds_off, const void* gptr,
                                            unsigned nElems) {
  unsigned long long ga = (unsigned long long)gptr;
  u32x4 g0;
  g0[0] = 1u;                                   // count=1, user descriptor
  g0[1] = lds_off;                              // LDS byte address
  g0[2] = (unsigned)ga;                         // global_addr[31:0]
  g0[3] = (unsigned)(ga >> 32) | (2u << 30);    // global_addr[56:32] | type=2
  i32x8 g1;
  g1[0] = (int)(2u << 16);                      // data_size = 2 (4 bytes)
  g1[1] = (int)((nElems & 0xFFFFu) << 16);      // tensor_dim0[15:0]
  g1[2] = (int)((nElems >> 16) | (1u << 16));   // tensor_dim0[31:16] | tensor_dim1=1
  g1[3] = (int)((nElems & 0xFFFFu) << 16);      // tile_dim0 (16-bit)
  g1[4] = 0;                                    // tile_dim1=0, tile_dim2=0 (1-D)
  g1[5] = (int)nElems;                          // tensor_dim0_stride[31:0]
  g1[6] = 0;
  g1[7] = 0;
  i32x4 z4 = {0, 0, 0, 0};
  i32x8 z8 = {0, 0, 0, 0, 0, 0, 0, 0};
  __builtin_amdgcn_tensor_load_to_lds(g0, g1, z4, z4, z8, 0);
}
#else
#define HAVE_TDM 0
#endif

// ---------------------------------------------------------------- WMMA GEMM
// out[row,col] = relu( sum_k mean[row,k]*Wl[k,col] + x[row,k]*Wr[k,col] + bl[col] )
// One block per 16-row strip (tm = blockIdx.x); wave w computes column tile w.
// A-panels (agg rows, x rows: 8KB each, contiguous) staged into LDS via TDM by
// wave 0, consumed by all waves -> 8x less A traffic than tile-per-wave.
// K = 128 in steps of 4 (V_WMMA_F32_16X16X4_F32); two WMMAs per step.
// A layout (ISA 7.12.2): lane<16 holds {K=k,k+1} of row lane&15; lane>=16 {k+2,k+3}.
// C/D: VGPR r -> row (r + 8*(lane>=16)), col (lane&15).
__global__ void k_sage_wmma(const float* __restrict__ agg,
                            const float* __restrict__ invdeg,
                            const float* __restrict__ xin,
                            const float* __restrict__ Wl,
                            const float* __restrict__ bl,
                            const float* __restrict__ Wr,
                            float* __restrict__ out,
                            int Hout) {
  __shared__ float ldsA[16 * D_FEAT];   // agg (pre-mean) panel
  __shared__ float ldsX[16 * D_FEAT];   // x panel

  const int wave = threadIdx.x >> 5;
  const int lane = threadIdx.x & 31;
  const int tm   = blockIdx.x;

  const float* aggPanel = agg + (size_t)tm * 16 * D_FEAT;
  const float* xPanel   = xin + (size_t)tm * 16 * D_FEAT;

#if HAVE_TDM
  if (wave == 0) {
    tdm_load_1d((unsigned)(unsigned long long)&ldsA[0], aggPanel, 16 * D_FEAT);
    tdm_load_1d((unsigned)(unsigned long long)&ldsX[0], xPanel,   16 * D_FEAT);
    __builtin_amdgcn_s_wait_tensorcnt(0);
  }
#else
  for (int i = threadIdx.x; i < 16 * D_FEAT; i += blockDim.x) {
    ldsA[i] = aggPanel[i];
    ldsX[i] = xPanel[i];
  }
#endif
  __syncthreads();

  const int tn      = wave;            // blockDim.x = 32 * nColTiles
  const int halfSel = lane >> 4;       // 0: K pair {k,k+1}, 1: {k+2,k+3}
  const int l15     = lane & 15;

  const int col  = tn * 16 + l15;
  const int colc = (col < Hout) ? col : (Hout - 1);   // keep B/bias loads in-bounds
  const float inv = invdeg[tm * 16 + l15];            // fold mean = agg/deg into A

  const float* aRow = &ldsA[l15 * D_FEAT];
  const float* xRow = &ldsX[l15 * D_FEAT];

  v8f acc = {};
  for (int k = 0; k < D_FEAT; k += 4) {
    const int kk = k + halfSel * 2;

    v2f a1; a1.x = aRow[kk] * inv; a1.y = aRow[kk + 1] * inv;
    v2f b1; b1.x = Wl[(size_t)kk * Hout + colc];
            b1.y = Wl[(size_t)(kk + 1) * Hout + colc];
    acc = __builtin_amdgcn_wmma_f32_16x16x4_f32(false, a1, false, b1,
                                                (short)0, acc, false, false);

    v2f a2; a2.x = xRow[kk]; a2.y = xRow[kk + 1];
    v2f b2; b2.x = Wr[(size_t)kk * Hout + colc];
            b2.y = Wr[(size_t)(kk + 1) * Hout + colc];
    acc = __builtin_amdgcn_wmma_f32_16x16x4_f32(false, a2, false, b2,
                                                (short)0, acc, false, false);
  }

  const float bias = bl[colc];
  if (col < Hout) {                    // data mask (layer 2 partial tile only)
#pragma unroll
    for (int r = 0; r < 8; ++r) {
      const int row = tm * 16 + halfSel * 8 + r;
      const float v = acc[r] + bias;
      out[(size_t)row * Hout + col] = fmaxf(v, 0.0f);
    }
  }
}

// ---------------------------------------------------------------- log_softmax
// One wave32 per row of 40 classes; lane handles cols {lane, lane+32}.
__global__ void k_logsoftmax(float* __restrict__ out, int nRows, int C) {
  const int wave = threadIdx.x >> 5;
  const int lane = threadIdx.x & 31;
  const int row  = blockIdx.x * (blockDim.x >> 5) + wave;
  if (row >= nRows) return;
  float* p = out + (size_t)row * C;

  const bool has2 = (lane + 32) < C;
  float v0 = p[lane];
  float v1 = has2 ? p[lane + 32] : -INFINITY;

  float m = fmaxf(v0, v1);
#pragma unroll
  for (int off = 16; off >= 1; off >>= 1)
    m = fmaxf(m, __shfl_xor(m, off, 32));

  float s = __expf(v0 - m) + (has2 ? __expf(v1 - m) : 0.0f);
#pragma unroll
  for (int off = 16; off >= 1; off >>= 1)
    s += __shfl_xor(s, off, 32);

  const float lse = m + __logf(s);
  p[lane] = v0 - lse;
  if (has2) p[lane + 32] = v1 - lse;
}

// ---------------------------------------------------------------- launch
extern "C" void kernel_launch(void* const* d_in, const int* in_sizes, int n_in,
                              void* d_out, int out_size, void* d_ws, size_t ws_size,
                              hipStream_t stream) {
  const float*     x   = (const float*)d_in[0];
  const long long* ei  = (const long long*)d_in[1];   // int64 (2, E) row-major
  const float*     Wl0 = (const float*)d_in[2];
  const float*     bl0 = (const float*)d_in[3];
  const float*     Wr0 = (const float*)d_in[4];
  const float*     Wl1 = (const float*)d_in[5];
  const float*     bl1 = (const float*)d_in[6];
  const float*     Wr1 = (const float*)d_in[7];
  const float*     Wl2 = (const float*)d_in[8];
  const float*     bl2 = (const float*)d_in[9];
  const float*     Wr2 = (const float*)d_in[10];
  float* outp = (float*)d_out;

  const long long* src = ei;
  const long long* dst = ei + N_EDGES;

  // Workspace layout (floats): deg[50048] | agg[N*128] | hA[N*128] | hB[N*128]
  float* deg = (float*)d_ws;
  float* agg = deg + 50048;
  float* hA  = agg + (size_t)N_NODES * D_FEAT;
  float* hB  = hA  + (size_t)N_NODES * D_FEAT;

  const int B = 256;
  const int nAgg   = N_NODES * D_FEAT;                  // 6,400,000
  const int gZero0 = (50048 + nAgg + B - 1) / B;
  const int gZero  = (nAgg + B - 1) / B;
  const int gDeg   = (N_EDGES + B - 1) / B;
  const int gInv   = (N_NODES + B - 1) / B;
  const int gScat  = (N_EDGES * 32) / B;                // exact: 100000
  const int gLsm   = (N_NODES + 3) / 4;                 // 4 rows/block (128 thr)

  // Degree (shared by all layers)
  k_zero<<<gZero0, B, 0, stream>>>(deg, 50048 + nAgg);
  k_degree<<<gDeg, B, 0, stream>>>(dst, deg);
  k_invdeg<<<gInv, B, 0, stream>>>(deg);

  // Layer 0: x -> hA     (8 col tiles -> 256 threads = 8 waves per strip)
  k_scatter<<<gScat, B, 0, stream>>>(x, src, dst, agg);
  k_sage_wmma<<<M_TILES, 256, 0, stream>>>(agg, deg, x, Wl0, bl0, Wr0, hA, HIDDEN);

  // Layer 1: hA -> hB
  k_zero<<<gZero, B, 0, stream>>>(agg, nAgg);
  k_scatter<<<gScat, B, 0, stream>>>(hA, src, dst, agg);
  k_sage_wmma<<<M_TILES, 256, 0, stream>>>(agg, deg, hA, Wl1, bl1, Wr1, hB, HIDDEN);

  // Layer 2: hB -> d_out (N x 40; 3 col tiles -> 96 threads = 3 waves per strip)
  k_zero<<<gZero, B, 0, stream>>>(agg, nAgg);
  k_scatter<<<gScat, B, 0, stream>>>(hB, src, dst, agg);
  k_sage_wmma<<<M_TILES, 96, 0, stream>>>(agg, deg, hB, Wl2, bl2, Wr2, outp, N_CLASSES);

  k_logsoftmax<<<gLsm, 128, 0, stream>>>(outp, N_NODES, N_CLASSES);
}